// StudentSSM_47356309406140
// MI455X (gfx1250) — compile-verified
//
#include <hip/hip_runtime.h>
#include <math.h>

// ---- problem constants ----
#define B_SZ     16
#define L_SEQ    2048
#define T_TOK    (B_SZ * L_SEQ)       // 32768 tokens
#define DMODEL   64
#define DINNER   128
#define DSTATE   16
#define DTRANK   4
#define NLAYERS  4
#define DCONV    4
#define NCLASSES 35
#define N_XZ     (2 * DINNER)          // 256
#define N_X2     (DINNER + 2 * DSTATE) // 160 (delta_pre | B | C)
#define NCH      32                    // scan chunks per sequence
#define CL       (L_SEQ / NCH)         // 64 tokens per chunk
#define SUB      16                    // BC LDS staging sub-chunk

typedef __attribute__((ext_vector_type(2))) float v2f;
typedef __attribute__((ext_vector_type(8))) float v8f;

__device__ __forceinline__ float silu_(float x) { return x / (1.f + __expf(-x)); }
__device__ __forceinline__ float softplus_(float x) {
    return (x > 20.f) ? x : __logf(1.f + __expf(x));
}

// D = A(16x4) * B(4x16) + C, fp32, wave32
__device__ __forceinline__ v8f wmma4(v2f a, v2f b, v8f c) {
    return __builtin_amdgcn_wmma_f32_16x16x4_f32(
        false, a, false, b, (short)0, c, false, false);
}

// ---- async global -> LDS (16B per lane), tracked by ASYNCcnt ----
__device__ __forceinline__ void async_copy16(const float* sdst, const float* gsrc) {
    unsigned lds_off = (unsigned)(unsigned long long)sdst; // low 32b of generic shared ptr = LDS byte offset
    asm volatile("global_load_async_to_lds_b128 %0, %1, off"
                 :: "v"(lds_off), "v"((unsigned long long)gsrc)
                 : "memory");
}
__device__ __forceinline__ void async_wait0() {
    asm volatile("s_wait_asynccnt 0" ::: "memory");
}

// ---------------- embed: h[t,d] = x[t]*W_proj[d] + b_proj[d] ----------------
__global__ void __launch_bounds__(256)
k_embed(const float* __restrict__ x, const float* __restrict__ wproj,
        const float* __restrict__ bproj, float* __restrict__ h) {
    int idx = blockIdx.x * 256 + threadIdx.x;
    int t = idx >> 6, d = idx & 63;
    h[idx] = x[t] * wproj[d] + bproj[d];
}

// ---------------- layernorm: one wave per token (64 elems) ----------------
__global__ void __launch_bounds__(256)
k_ln(const float* __restrict__ h, const float* __restrict__ g,
     const float* __restrict__ bt, float* __restrict__ out) {
    int lane = threadIdx.x & 31, wid = threadIdx.x >> 5;
    int t = blockIdx.x * 8 + wid;
    const float* hp = h + t * DMODEL;
    float x0 = hp[lane], x1 = hp[lane + 32];
    float s = x0 + x1, s2 = x0 * x0 + x1 * x1;
#pragma unroll
    for (int off = 16; off; off >>= 1) {
        s  += __shfl_xor(s,  off, 32);
        s2 += __shfl_xor(s2, off, 32);
    }
    float mu  = s * (1.f / 64.f);
    float var = s2 * (1.f / 64.f) - mu * mu;
    float inv = rsqrtf(var + 1e-5f);
    float* op = out + t * DMODEL;
    op[lane]      = (x0 - mu) * inv * g[lane]      + bt[lane];
    op[lane + 32] = (x1 - mu) * inv * g[lane + 32] + bt[lane + 32];
}

// -------- GEMM1: xz = ln(T x 64) @ W_in^T(64 x 256); split -> uc | z --------
// W (256x64) staged in LDS via async copy; row stride padded to 68 (4r banks).
#define SWS1 (DMODEL + 4)
__global__ void __launch_bounds__(256)
k_gemm_in(const float* __restrict__ A, const float* __restrict__ W,
          float* __restrict__ uc, float* __restrict__ z) {
    __shared__ float sW[N_XZ * SWS1];
    const int tid = threadIdx.x;
    for (int c = tid; c < N_XZ * (DMODEL / 4); c += 256) {
        int row = c >> 4, cp = (c & 15) << 2;
        async_copy16(&sW[row * SWS1 + cp], W + row * DMODEL + cp);
    }
    async_wait0();
    __syncthreads();

    const int lane = tid & 31;
    const int wid  = tid >> 5;
    const int rt   = blockIdx.x * 8 + wid;       // row tile (16 tokens)
    const int r15  = lane & 15;
    const int kb   = (lane >> 4) << 1;
    const float* ap = A + (rt * 16 + r15) * DMODEL + kb;

    v2f areg[16];
#pragma unroll
    for (int i = 0; i < 16; ++i) areg[i] = *(const v2f*)(ap + 4 * i);

    const int n  = lane & 15;
    const int mh = (lane >> 4) << 3;

    for (int cg = 0; cg < 4; ++cg) {             // 4 groups of 64 cols
        const int n0 = cg * 64;
        const float* wp = &sW[(n0 + r15) * SWS1 + kb];
        v8f c0 = {}, c1 = {}, c2 = {}, c3 = {};
#pragma unroll
        for (int kk = 0; kk < DMODEL; kk += 4) {
            v2f a  = areg[kk >> 2];
            v2f b0 = *(const v2f*)(wp + kk);
            v2f b1 = *(const v2f*)(wp + 16 * SWS1 + kk);
            v2f b2 = *(const v2f*)(wp + 32 * SWS1 + kk);
            v2f b3 = *(const v2f*)(wp + 48 * SWS1 + kk);
            c0 = wmma4(a, b0, c0);
            c1 = wmma4(a, b1, c1);
            c2 = wmma4(a, b2, c2);
            c3 = wmma4(a, b3, c3);
        }
        v8f cc[4] = {c0, c1, c2, c3};
#pragma unroll
        for (int tix = 0; tix < 4; ++tix) {
            int ng = n0 + tix * 16 + n;
            float* dst = (ng < DINNER) ? uc : z;
            int col    = (ng < DINNER) ? ng : (ng - DINNER);
#pragma unroll
            for (int rr = 0; rr < 8; ++rr)
                dst[(rt * 16 + mh + rr) * DINNER + col] = cc[tix][rr];
        }
    }
}

// -------- causal depthwise conv (k=4) + bias + silu: uc -> u --------
__global__ void __launch_bounds__(256)
k_conv(const float* __restrict__ uc, const float* __restrict__ w,
       const float* __restrict__ cb, float* __restrict__ u) {
    int idx = blockIdx.x * 256 + threadIdx.x;
    int t = idx >> 7, e = idx & 127;
    int l = t & (L_SEQ - 1);
    float acc = cb[e];
#pragma unroll
    for (int j = 0; j < DCONV; ++j) {
        int lo = l - (DCONV - 1) + j;
        if (lo >= 0) acc += w[e * DCONV + j] * uc[(t - (DCONV - 1) + j) * DINNER + e];
    }
    u[idx] = silu_(acc);
}

// -------- prep: Wxc(160x128/layer) = [W_dt @ W_x[:4] ; W_x[4:36]] --------
__global__ void __launch_bounds__(256)
k_prep_wxc(const float* __restrict__ W_x, const float* __restrict__ W_dt,
           float* __restrict__ wxc) {
    int idx = blockIdx.x * 256 + threadIdx.x;
    if (idx >= NLAYERS * N_X2 * DINNER) return;
    int l = idx / (N_X2 * DINNER);
    int rem = idx - l * (N_X2 * DINNER);
    int row = rem / DINNER;
    int j = rem - row * DINNER;
    const float* wx = W_x + l * ((DTRANK + 2 * DSTATE) * DINNER);
    float v;
    if (row < DINNER) {
        const float* wdt = W_dt + l * (DINNER * DTRANK) + row * DTRANK;
        v = 0.f;
#pragma unroll
        for (int q = 0; q < DTRANK; ++q) v += wdt[q] * wx[q * DINNER + j];
    } else {
        v = wx[(DTRANK + (row - DINNER)) * DINNER + j];
    }
    wxc[idx] = v;
}

// -------- GEMM2: u(T x 128) @ Wxc^T(128 x 160) -> delta(T x 128), bc(T x 32) ----
#define SWS2 (DINNER + 4)
__global__ void __launch_bounds__(256)
k_gemm_x(const float* __restrict__ A, const float* __restrict__ W,
         const float* __restrict__ bdt, float* __restrict__ delta,
         float* __restrict__ bc) {
    __shared__ float sW[N_X2 * SWS2];
    const int tid = threadIdx.x;
    for (int c = tid; c < N_X2 * (DINNER / 4); c += 256) {
        int row = c >> 5, cp = (c & 31) << 2;
        async_copy16(&sW[row * SWS2 + cp], W + row * DINNER + cp);
    }
    async_wait0();
    __syncthreads();

    const int lane = tid & 31;
    const int wid  = tid >> 5;
    const int rt   = blockIdx.x * 8 + wid;
    const int r15  = lane & 15;
    const int kb   = (lane >> 4) << 1;
    const float* ap = A + (rt * 16 + r15) * DINNER + kb;

    v2f areg[32];
#pragma unroll
    for (int i = 0; i < 32; ++i) areg[i] = *(const v2f*)(ap + 4 * i);

    const int n  = lane & 15;
    const int mh = (lane >> 4) << 3;

    for (int cg = 0; cg < 2; ++cg) {             // 2 groups of 5 tiles (80 cols)
        const int n0 = cg * 80;
        const float* wp[5];
#pragma unroll
        for (int tix = 0; tix < 5; ++tix)
            wp[tix] = &sW[(n0 + tix * 16 + r15) * SWS2 + kb];
        v8f c[5] = {};
#pragma unroll
        for (int kk = 0; kk < DINNER; kk += 4) {
            v2f a = areg[kk >> 2];
#pragma unroll
            for (int tix = 0; tix < 5; ++tix) {
                v2f b = *(const v2f*)(wp[tix] + kk);
                c[tix] = wmma4(a, b, c[tix]);
            }
        }
#pragma unroll
        for (int tix = 0; tix < 5; ++tix) {
            int ng = n0 + tix * 16 + n;
            if (ng < DINNER) {
                float bias = bdt[ng];
#pragma unroll
                for (int rr = 0; rr < 8; ++rr)
                    delta[(rt * 16 + mh + rr) * DINNER + ng] = softplus_(c[tix][rr] + bias);
            } else {
#pragma unroll
                for (int rr = 0; rr < 8; ++rr)
                    bc[(rt * 16 + mh + rr) * 32 + (ng - DINNER)] = c[tix][rr];
            }
        }
    }
}

// ======================= chunked parallel scan =======================
// Pass A: per-chunk local scan (zero init): end-state hend and decay product P.
__global__ void __launch_bounds__(128)
k_scan_local(const float* __restrict__ delta, const float* __restrict__ u,
             const float* __restrict__ bc, const float* __restrict__ A_log,
             float* __restrict__ hend, float* __restrict__ Pb) {
    const int b  = blockIdx.x / NCH;
    const int ch = blockIdx.x - b * NCH;
    const int e  = threadIdx.x;
    const int t0 = b * L_SEQ + ch * CL;
    __shared__ float sBC[SUB][32];

    float Av[DSTATE], hs[DSTATE], P[DSTATE];
#pragma unroll
    for (int nn = 0; nn < DSTATE; ++nn) {
        Av[nn] = -__expf(A_log[e * DSTATE + nn]);
        hs[nn] = 0.f;
        P[nn]  = 1.f;
    }
    for (int l0 = 0; l0 < CL; l0 += SUB) {
        __syncthreads();
        for (int i = threadIdx.x; i < SUB * 32; i += 128)
            sBC[i >> 5][i & 31] = bc[(t0 + l0 + (i >> 5)) * 32 + (i & 31)];
        __syncthreads();
#pragma unroll 4
        for (int li = 0; li < SUB; ++li) {
            int t = t0 + l0 + li;
            float d  = delta[t * DINNER + e];
            float du = d * u[t * DINNER + e];
#pragma unroll
            for (int nn = 0; nn < DSTATE; ++nn) {
                float a = __expf(d * Av[nn]);
                hs[nn] = a * hs[nn] + du * sBC[li][nn];
                P[nn] *= a;
            }
        }
    }
    float* hp = hend + ((size_t)(b * NCH + ch) * DINNER + e) * DSTATE;
    float* pp = Pb   + ((size_t)(b * NCH + ch) * DINNER + e) * DSTATE;
#pragma unroll
    for (int nn = 0; nn < DSTATE; ++nn) { hp[nn] = hs[nn]; pp[nn] = P[nn]; }
}

// Pass B: serial combine over chunks: s[ch+1] = P[ch]*s[ch] + hend[ch].
__global__ void __launch_bounds__(256)
k_scan_combine(const float* __restrict__ hend, const float* __restrict__ Pb,
               float* __restrict__ s0) {
    int idx = blockIdx.x * 256 + threadIdx.x;     // over B*DINNER = 2048
    if (idx >= B_SZ * DINNER) return;
    int b = idx >> 7, e = idx & 127;
    float s[DSTATE];
#pragma unroll
    for (int nn = 0; nn < DSTATE; ++nn) s[nn] = 0.f;
    for (int ch = 0; ch < NCH; ++ch) {
        size_t base = ((size_t)(b * NCH + ch) * DINNER + e) * DSTATE;
#pragma unroll
        for (int nn = 0; nn < DSTATE; ++nn) s0[base + nn] = s[nn];
#pragma unroll
        for (int nn = 0; nn < DSTATE; ++nn)
            s[nn] = Pb[base + nn] * s[nn] + hend[base + nn];
    }
}

// Pass C: rescan each chunk from corrected init; compute y, D skip, z gating.
__global__ void __launch_bounds__(128)
k_scan_fix(const float* __restrict__ delta, const float* __restrict__ u,
           const float* __restrict__ bc, const float* __restrict__ z,
           const float* __restrict__ A_log, const float* __restrict__ Dp,
           const float* __restrict__ s0, float* __restrict__ yv) {
    const int b  = blockIdx.x / NCH;
    const int ch = blockIdx.x - b * NCH;
    const int e  = threadIdx.x;
    const int t0 = b * L_SEQ + ch * CL;
    __shared__ float sBC[SUB][32];

    float Av[DSTATE], hs[DSTATE];
    const float* sp = s0 + ((size_t)(b * NCH + ch) * DINNER + e) * DSTATE;
#pragma unroll
    for (int nn = 0; nn < DSTATE; ++nn) {
        Av[nn] = -__expf(A_log[e * DSTATE + nn]);
        hs[nn] = sp[nn];
    }
    const float dpe = Dp[e];

    for (int l0 = 0; l0 < CL; l0 += SUB) {
        __syncthreads();
        for (int i = threadIdx.x; i < SUB * 32; i += 128)
            sBC[i >> 5][i & 31] = bc[(t0 + l0 + (i >> 5)) * 32 + (i & 31)];
        __syncthreads();
#pragma unroll 4
        for (int li = 0; li < SUB; ++li) {
            int t = t0 + l0 + li;
            float d  = delta[t * DINNER + e];
            float ut = u[t * DINNER + e];
            float du = d * ut;
            float y = 0.f;
#pragma unroll
            for (int nn = 0; nn < DSTATE; ++nn) {
                hs[nn] = __expf(d * Av[nn]) * hs[nn] + du * sBC[li][nn];
                y += hs[nn] * sBC[li][DSTATE + nn];
            }
            float zt = z[t * DINNER + e];
            yv[t * DINNER + e] = (y + ut * dpe) * silu_(zt);
        }
    }
}

// -------- GEMM3: h += yv(T x 128) @ W_out^T(128 x 64) --------
__global__ void __launch_bounds__(256)
k_gemm_out(const float* __restrict__ A, const float* __restrict__ W,
           float* __restrict__ h) {
    __shared__ float sW[DMODEL * SWS2];
    const int tid = threadIdx.x;
    for (int c = tid; c < DMODEL * (DINNER / 4); c += 256) {
        int row = c >> 5, cp = (c & 31) << 2;
        async_copy16(&sW[row * SWS2 + cp], W + row * DINNER + cp);
    }
    async_wait0();
    __syncthreads();

    const int lane = tid & 31;
    const int wid  = tid >> 5;
    const int rt   = blockIdx.x * 8 + wid;
    const int r15  = lane & 15;
    const int kb   = (lane >> 4) << 1;
    const float* ap = A + (rt * 16 + r15) * DINNER + kb;
    const float* wp = &sW[r15 * SWS2 + kb];

    v8f c0 = {}, c1 = {}, c2 = {}, c3 = {};
#pragma unroll
    for (int kk = 0; kk < DINNER; kk += 4) {
        v2f a  = *(const v2f*)(ap + kk);
        v2f b0 = *(const v2f*)(wp + kk);
        v2f b1 = *(const v2f*)(wp + 16 * SWS2 + kk);
        v2f b2 = *(const v2f*)(wp + 32 * SWS2 + kk);
        v2f b3 = *(const v2f*)(wp + 48 * SWS2 + kk);
        c0 = wmma4(a, b0, c0);
        c1 = wmma4(a, b1, c1);
        c2 = wmma4(a, b2, c2);
        c3 = wmma4(a, b3, c3);
    }
    const int n  = lane & 15;
    const int mh = (lane >> 4) << 3;
    v8f cc[4] = {c0, c1, c2, c3};
#pragma unroll
    for (int tix = 0; tix < 4; ++tix) {
        int ng = tix * 16 + n;
#pragma unroll
        for (int rr = 0; rr < 8; ++rr)
            h[(rt * 16 + mh + rr) * DMODEL + ng] += cc[tix][rr];
    }
}

// -------- head: mean over L then (B x 64) @ W_head^T + b_head --------
__global__ void __launch_bounds__(64)
k_head(const float* __restrict__ h, const float* __restrict__ Wh,
       const float* __restrict__ bh, float* __restrict__ out) {
    const int b = blockIdx.x;
    const int d = threadIdx.x;
    __shared__ float hm[DMODEL];
    float s = 0.f;
    for (int l = 0; l < L_SEQ; ++l) s += h[(b * L_SEQ + l) * DMODEL + d];
    hm[d] = s * (1.f / (float)L_SEQ);
    __syncthreads();
    if (d < NCLASSES) {
        float o = bh[d];
#pragma unroll
        for (int k = 0; k < DMODEL; ++k) o += hm[k] * Wh[d * DMODEL + k];
        out[b * NCLASSES + d] = o;
    }
}

extern "C" void kernel_launch(void* const* d_in, const int* in_sizes, int n_in,
                              void* d_out, int out_size, void* d_ws, size_t ws_size,
                              hipStream_t stream) {
    const float* x      = (const float*)d_in[0];
    const float* W_proj = (const float*)d_in[1];
    const float* b_proj = (const float*)d_in[2];
    const float* ln_g   = (const float*)d_in[3];
    const float* ln_b   = (const float*)d_in[4];
    const float* W_in   = (const float*)d_in[5];
    const float* conv_w = (const float*)d_in[6];
    const float* conv_b = (const float*)d_in[7];
    const float* W_x    = (const float*)d_in[8];
    const float* W_dt   = (const float*)d_in[9];
    const float* b_dt   = (const float*)d_in[10];
    const float* A_log  = (const float*)d_in[11];
    const float* Dp     = (const float*)d_in[12];
    const float* W_out  = (const float*)d_in[13];
    const float* W_head = (const float*)d_in[14];
    const float* b_head = (const float*)d_in[15];
    float* out = (float*)d_out;

    float* ws   = (float*)d_ws;
    float* h    = ws;                            // T*64
    float* ln   = h   + (size_t)T_TOK * DMODEL;  // T*64
    float* uc   = ln  + (size_t)T_TOK * DMODEL;  // T*128 (reused as yv after conv)
    float* u    = uc  + (size_t)T_TOK * DINNER;  // T*128
    float* z    = u   + (size_t)T_TOK * DINNER;  // T*128
    float* dl   = z   + (size_t)T_TOK * DINNER;  // T*128
    float* bcb  = dl  + (size_t)T_TOK * DINNER;  // T*32
    float* wxc  = bcb + (size_t)T_TOK * 32;      // 4*160*128
    float* hend = wxc + (size_t)NLAYERS * N_X2 * DINNER; // B*NCH*128*16
    float* Pb   = hend + (size_t)B_SZ * NCH * DINNER * DSTATE;
    float* s0   = Pb   + (size_t)B_SZ * NCH * DINNER * DSTATE;
    float* yv   = uc;                            // alias: uc free after conv

    k_embed<<<(T_TOK * DMODEL) / 256, 256, 0, stream>>>(x, W_proj, b_proj, h);
    k_prep_wxc<<<(NLAYERS * N_X2 * DINNER + 255) / 256, 256, 0, stream>>>(W_x, W_dt, wxc);

    for (int layer = 0; layer < NLAYERS; ++layer) {
        k_ln<<<T_TOK / 8, 256, 0, stream>>>(h, ln_g + layer * DMODEL,
                                            ln_b + layer * DMODEL, ln);
        k_gemm_in<<<T_TOK / 128, 256, 0, stream>>>(ln, W_in + layer * N_XZ * DMODEL, uc, z);
        k_conv<<<(T_TOK * DINNER) / 256, 256, 0, stream>>>(
            uc, conv_w + layer * DINNER * DCONV, conv_b + layer * DINNER, u);
        k_gemm_x<<<T_TOK / 128, 256, 0, stream>>>(
            u, wxc + layer * N_X2 * DINNER, b_dt + layer * DINNER, dl, bcb);
        k_scan_local<<<B_SZ * NCH, DINNER, 0, stream>>>(
            dl, u, bcb, A_log + layer * DINNER * DSTATE, hend, Pb);
        k_scan_combine<<<(B_SZ * DINNER + 255) / 256, 256, 0, stream>>>(hend, Pb, s0);
        k_scan_fix<<<B_SZ * NCH, DINNER, 0, stream>>>(
            dl, u, bcb, z, A_log + layer * DINNER * DSTATE, Dp + layer * DINNER, s0, yv);
        k_gemm_out<<<T_TOK / 128, 256, 0, stream>>>(
            yv, W_out + layer * DMODEL * DINNER, h);
    }
    k_head<<<B_SZ, DMODEL, 0, stream>>>(h, W_head, b_head, out);
}